// SfmPool_62251255988801
// MI455X (gfx1250) — compile-verified
//
#include <hip/hip_runtime.h>
#include <math.h>

// SfmPool fused forward for MI455X (gfx1250).
// Bandwidth-bound: ~504 MB min HBM traffic -> single fused pass.
// CDNA5 Tensor Data Mover (tensor_load_to_lds) DMAs a 3D strided tile
// (8 ch x 128 cols x 7 rows) global->LDS; then window maxima -> max_v
// (sum of covering-window maxima, incl. overlap counts) -> safe_exp ->
// pooled num/den, one output write per element.
//
// Grid is 3-D (cchunk, oh, b) so no integer division in the prologue.

constexpr int H  = 128;
constexpr int W  = 128;
constexpr int C  = 192;
constexpr int OH = 64;
constexpr int OW = 64;
constexpr int CC = 8;            // channels per workgroup
constexpr int NCH = C / CC;      // 24 chunks
constexpr int NTHREADS = 512;    // 64 output cols x 8 channels

typedef __attribute__((ext_vector_type(4))) unsigned int u32x4_t;
typedef __attribute__((ext_vector_type(8))) int          i32x8_t;
typedef __attribute__((ext_vector_type(4))) int          i32x4_t;

__global__ __launch_bounds__(NTHREADS)
void sfmpool_kernel(const float* __restrict__ in, float* __restrict__ out) {
    __shared__ __align__(16) float s_in[7][W][CC];   // 28672 B raw tile
    __shared__ float s_hmax[7][OW][CC];              // 14336 B horizontal 3-max
    __shared__ float s_wmax[3][OW][CC];              //  6144 B window maxima wr=oh-1..oh+1

    const int tid = threadIdx.x;
    const int c0  = (int)blockIdx.x * CC;            // channel chunk
    const int oh  = (int)blockIdx.y;                 // output row
    const int b   = (int)blockIdx.z;                 // batch

    // Rows needed: 2oh-2 .. 2oh+4; clamp start so all 7 loaded rows are real
    // (padding rows handled by masking below, matching JAX -inf/0 pads).
    int r0 = 2 * oh - 2;
    r0 = r0 < 0 ? 0 : r0;
    r0 = r0 > (H - 7) ? (H - 7) : r0;

    const float* gsrc = in + ((size_t)(b * H + r0) * W) * C + c0;

#if __has_builtin(__builtin_amdgcn_tensor_load_to_lds)
    // Tensor Data Mover: 3D tile, x = 8 contiguous channels, y = 128 cols
    // (stride C), z = 7 rows (stride W*C). Only wave 0 issues the DMA
    // (TDM ignores EXEC; the scalar branch keeps other waves from
    // duplicating it). Wave 0 drains TENSORcnt before the barrier, so the
    // barrier release implies the tile has landed in LDS.
    if (tid < 32) {
        unsigned long long ga = (unsigned long long)(const void*)gsrc;
        unsigned int lds_off  = (unsigned int)(unsigned long long)(const void*)&s_in[0][0][0];

        u32x4_t g0;                          // D# group 0
        g0[0] = 1u;                          // count=1 (valid), user mode, no gather
        g0[1] = lds_off;                     // lds_addr (bytes, workgroup-relative)
        g0[2] = (unsigned int)ga;            // global_addr[31:0]
        g0[3] = (unsigned int)((ga >> 32) & 0x01FFFFFFull) | (2u << 30); // addr[56:32] | type=2

        i32x8_t g1;                          // D# group 1
        g1[0] = (int)(2u << 16);             // data_size=2 (4 bytes)
        g1[1] = (int)(192u << 16);           // tensor_dim0 lo16 (=C) at bits[79:48]
        g1[2] = (int)(128u << 16);           // tensor_dim0 hi=0 | tensor_dim1 lo = 128
        g1[3] = (int)((unsigned)CC << 16);   // tensor_dim1 hi=0 | tile_dim0 = 8
        g1[4] = (int)((7u << 16) | 128u);    // tile_dim1 = 128 | tile_dim2 = 7
        g1[5] = 192;                         // tensor_dim0_stride = C (col step)
        g1[6] = (int)((24576u & 0xFFFFu) << 16); // stride0 hi=0 | tensor_dim1_stride lo (W*C)
        g1[7] = (int)(24576u >> 16);         // tensor_dim1_stride hi

        i32x4_t g2;                          // D# group 2
        g2[0] = H;                           // tensor_dim2 = 128 (z in [0,7) never OOB)
        g2[1] = 1;                           // tensor_dim3 = 1 (safety; tile_dim3 unused)
        g2[2] = 0;                           // tensor_dim2_stride lo (no 4th dim)
        g2[3] = 0;                           // stride hi | tile_dim3 = 0 (unused)

        i32x4_t g3;                          // D# group 3
        g3[0] = 0;                           // tensor_dim3_stride lo
        g3[1] = (int)(1u << 16);             // stride hi=0 | tensor_dim4 = 1 (safety)
        g3[2] = 0;                           // tensor_dim4 hi | tile_dim4 = 0
        g3[3] = 0;

#if defined(__clang_major__) && (__clang_major__ >= 23)
        i32x8_t gz = {0, 0, 0, 0, 0, 0, 0, 0};
        __builtin_amdgcn_tensor_load_to_lds(g0, g1, g2, g3, gz, 0);
#else
        __builtin_amdgcn_tensor_load_to_lds(g0, g1, g2, g3, 0);
#endif
        __builtin_amdgcn_s_wait_tensorcnt(0);
    }
#else
    // Fallback: cooperative plain loads.
    for (int idx = tid; idx < 7 * W * CC; idx += NTHREADS) {
        int c = idx % CC;
        int j = (idx / CC) % W;
        int r = idx / (CC * W);
        s_in[r][j][c] = gsrc[((size_t)r * W + j) * C + c];
    }
#endif
    __syncthreads();

    // Horizontal 3-max per loaded row: hmax[r][wc] = max_j s_in[r][2wc..2wc+2].
    // Column 128 is SAME-padding (-inf for max) -> masked.
    for (int idx = tid; idx < 7 * OW * CC; idx += NTHREADS) {
        int c   = idx % CC;
        int wcx = (idx / CC) % OW;
        int r   = idx / (CC * OW);
        int j0  = 2 * wcx;
        float m = fmaxf(s_in[r][j0][c], s_in[r][j0 + 1][c]);
        if (j0 + 2 < W) m = fmaxf(m, s_in[r][j0 + 2][c]);
        s_hmax[r][wcx][c] = m;
    }
    __syncthreads();

    // Vertical 3-max -> full window maxima for wr in {oh-1, oh, oh+1}.
    // Row 128 is padding -> masked; invalid wr slots hold -FLT_MAX (never read).
    for (int idx = tid; idx < 3 * OW * CC; idx += NTHREADS) {
        int c   = idx % CC;
        int wcx = (idx / CC) % OW;
        int a   = idx / (CC * OW);
        int wr  = oh - 1 + a;
        float m = -3.402823466e38f;
        if (wr >= 0 && wr < OH) {
            int i0 = 2 * wr;
            m = fmaxf(s_hmax[i0 - r0][wcx][c], s_hmax[i0 + 1 - r0][wcx][c]);
            if (i0 + 2 < H) m = fmaxf(m, s_hmax[i0 + 2 - r0][wcx][c]);
        }
        s_wmax[a][wcx][c] = m;
    }
    __syncthreads();

    // One output per thread: (oh, wc, c0+c).
    {
        int c  = tid % CC;
        int wc = tid / CC;
        float num = 0.f, den = 0.f;
        for (int di = 0; di < 3; ++di) {
            int i = 2 * oh + di;
            if (i >= H) break;               // pad rows contribute 0 to sums
            int aLo = (di == 0) ? 0 : 1;     // row windows covering element i
            int aHi = (di == 2) ? 2 : 1;
            for (int dj = 0; dj < 3; ++dj) {
                int j = 2 * wc + dj;
                if (j >= W) break;           // pad cols contribute 0 to sums
                int bLo = (dj == 0) ? (wc - 1) : wc;  // col windows covering j
                int bHi = (dj == 2) ? (wc + 1) : wc;
                float x  = s_in[i - r0][j][c];
                float mv = 0.f;              // SUM of covering-window maxima
                for (int a = aLo; a <= aHi; ++a) {
                    int wr = oh - 1 + a;
                    if (wr < 0 || wr >= OH) continue;
                    for (int wb = bLo; wb <= bHi; ++wb) {
                        if (wb < 0 || wb >= OW) continue;
                        mv += s_wmax[a][wb][c];
                    }
                }
                float se = expf(x - mv);
                num += x * se;
                den += se;
            }
        }
        out[((size_t)(b * OH + oh) * OW + wc) * C + c0 + c] = num / den;
    }
}

extern "C" void kernel_launch(void* const* d_in, const int* in_sizes, int n_in,
                              void* d_out, int out_size, void* d_ws, size_t ws_size,
                              hipStream_t stream) {
    (void)n_in; (void)out_size; (void)d_ws; (void)ws_size;
    const float* in = (const float*)d_in[0];
    float* out = (float*)d_out;
    const int N = in_sizes[0] / (H * W * C);   // 32
    dim3 grid((unsigned)NCH, (unsigned)OH, (unsigned)N);  // (24, 64, 32)
    sfmpool_kernel<<<grid, NTHREADS, 0, stream>>>(in, out);
}